// RelationEncoder_88476326297844
// MI455X (gfx1250) — compile-verified
//
#include <hip/hip_runtime.h>
#include <hip/hip_bf16.h>

typedef __attribute__((ext_vector_type(2))) float v2f;
typedef __attribute__((ext_vector_type(8))) float v8f;

#define M_TOK   262144          // B*T*N = 8*128*256
#define NTILES  (M_TOK / 16)    // 16384 row-tiles of 16 tokens
#define CIN     15
#define DIM     128
#define LDA     132             // h'-stage stride: banks (4m+k) -> conflict-free
#define LW1     136             // W1/LDS stride: banks (8k+n); k vs k+2 rows disjoint
#define BN_EPS  1e-5f

// ---------------------------------------------------------------------------
// V_WMMA_F32_16X16X4_F32: D(16x16,f32) = A(16x4) * B(4x16) + C
// (neg_a, A, neg_b, B, c_mod, C, reuse_a, reuse_b)
// ---------------------------------------------------------------------------
__device__ __forceinline__ v8f wmma_f32(v2f a, v2f b, v8f c) {
    return __builtin_amdgcn_wmma_f32_16x16x4_f32(false, a, false, b,
                                                 (short)0, c, false, false);
}

// ---------------------------------------------------------------------------
// One 16-row x 128-col tile of h = diff @ W1 + b1.
// W1 comes zero-padded (16 rows x LW1 stride) from LDS -> no guards.
// A fragment: lane<16 -> K={0,1}, lane>=16 -> K={2,3}; row = lane&15.
// Boundary element k=15 handled with clamped load + select (no exec branch).
// ---------------------------------------------------------------------------
__device__ __forceinline__ void gemm1_tile(const float* __restrict__ diff,
                                           const float* __restrict__ sW1,
                                           const float* __restrict__ b1,
                                           int row0, int lane, v8f c[8]) {
    const int m  = lane & 15;
    const int kb = (lane >> 4) << 1;                 // 0 or 2
    const float* drow = diff + (size_t)(row0 + m) * CIN;

    v2f a[4];
#pragma unroll
    for (int s = 0; s < 4; ++s) {
        const int k0 = 4 * s + kb;                   // <= 14 always
        const int k1 = k0 + 1;                       // == 15 only for s=3,kb=2
        a[s].x = drow[k0];
        float ty = drow[(k1 < CIN) ? k1 : (CIN - 1)];   // always in-bounds
        a[s].y = (k1 < CIN) ? ty : 0.f;                 // v_cndmask, no branch
    }
#pragma unroll
    for (int nt = 0; nt < 8; ++nt) {
        const int n = nt * 16 + m;
        const float bias = b1[n];
        v8f acc;
#pragma unroll
        for (int i = 0; i < 8; ++i) acc[i] = bias;   // b1 depends only on N
#pragma unroll
        for (int s = 0; s < 4; ++s) {
            const int k0 = 4 * s + kb;
            v2f b;
            b.x = sW1[k0 * LW1 + n];                 // row 15 is zero-padded
            b.y = sW1[(k0 + 1) * LW1 + n];
            acc = wmma_f32(a[s], b, acc);
        }
        c[nt] = acc;
    }
}

// Load W1 [15,128] into LDS zero-padded to [16, LW1].
template <int NT>
__device__ __forceinline__ void load_W1(const float* __restrict__ W1,
                                        float* __restrict__ sW1, int tid) {
    for (int i = tid; i < 16 * LW1; i += NT) {
        const int r = i / LW1, col = i - r * LW1;
        sW1[i] = (r < CIN && col < DIM) ? W1[r * DIM + col] : 0.f;
    }
}

// ---------------------------------------------------------------------------
// Phase 0: zero stats accumulators.
// ws layout (floats): [0..127]=sum [128..255]=sumsq [256]=count
//                     [257..384]=scale s [385..512]=shift t
// ---------------------------------------------------------------------------
__global__ void k_zero(float* __restrict__ ws) {
    int i = threadIdx.x;
    if (i < 257) ws[i] = 0.f;
}

// ---------------------------------------------------------------------------
// Phase 1: GEMM1 + masked per-channel sum / sumsq / count.
// 256 blocks x 8 waves, 8 tiles per wave.
// ---------------------------------------------------------------------------
__global__ __launch_bounds__(256)
void k_gemm1_stats(const float* __restrict__ diff, const int* __restrict__ mask,
                   const float* __restrict__ W1,   const float* __restrict__ b1,
                   float* __restrict__ ws) {
    __shared__ float sW1[16 * LW1];
    __shared__ float s_sum[DIM];
    __shared__ float s_ssq[DIM];
    __shared__ float s_cnt;

    const int tid  = threadIdx.x;
    const int wave = tid >> 5;
    const int lane = tid & 31;
    const int m    = lane & 15;

    load_W1<256>(W1, sW1, tid);
    if (tid < DIM) { s_sum[tid] = 0.f; s_ssq[tid] = 0.f; }
    if (tid == 0)  s_cnt = 0.f;
    __syncthreads();

    for (int t = blockIdx.x * 8 + wave; t < NTILES; t += 256 * 8) {
        const int row0 = t * 16;
        v8f c[8];
        gemm1_tile(diff, sW1, b1, row0, lane, c);

        const int rbase = row0 + ((lane >= 16) ? 8 : 0);
        float msk[8];
#pragma unroll
        for (int r = 0; r < 8; ++r) msk[r] = mask[rbase + r] ? 1.f : 0.f;

#pragma unroll
        for (int nt = 0; nt < 8; ++nt) {
            float ls = 0.f, lq = 0.f;
#pragma unroll
            for (int r = 0; r < 8; ++r) {
                float v = c[nt][r] * msk[r];      // h*m (m in {0,1})
                ls += v;
                lq += v * v;                      // == h^2 * m
            }
            ls += __shfl_xor(ls, 16, 32);         // fold the two half-waves
            lq += __shfl_xor(lq, 16, 32);
            if (lane < 16) {
                atomicAdd(&s_sum[nt * 16 + m], ls);   // ds_add_f32
                atomicAdd(&s_ssq[nt * 16 + m], lq);
            }
        }
        if (lane < 16) atomicAdd(&s_cnt, mask[row0 + lane] ? 1.f : 0.f);
    }
    __syncthreads();
    if (tid < DIM) {
        atomicAdd(&ws[tid],       s_sum[tid]);        // global_atomic_add_f32
        atomicAdd(&ws[DIM + tid], s_ssq[tid]);
    }
    if (tid == 0) atomicAdd(&ws[256], s_cnt);
}

// ---------------------------------------------------------------------------
// Phase 2: fold BN into per-channel affine:  y = h*s + t
// ---------------------------------------------------------------------------
__global__ void k_finalize(const float* __restrict__ gamma,
                           const float* __restrict__ beta,
                           float* __restrict__ ws) {
    int d = threadIdx.x;
    if (d >= DIM) return;
    float cnt  = fmaxf(ws[256], 1.f);
    float mean = ws[d] / cnt;
    float var  = ws[DIM + d] / cnt - mean * mean;     // masked E[h^2]-mean^2
    float inv  = rsqrtf(fmaxf(var, 0.f) + BN_EPS);
    float s    = gamma[d] * inv;
    ws[257 + d] = s;
    ws[385 + d] = beta[d] - mean * s;
}

// ---------------------------------------------------------------------------
// Phase 3: recompute h, BN+ReLU into LDS, then block-cooperative GEMM2.
// Block = 4 waves = 64 rows.  Each wave computes h' for its own 16-row tile,
// then owns 2 N-tiles across all 64 rows so each W2 fragment load feeds
// 4 WMMAs (4x less L2 traffic on W2).
// ---------------------------------------------------------------------------
__global__ __launch_bounds__(128)
void k_gemm2(const float* __restrict__ diff, const int* __restrict__ mask,
             const float* __restrict__ W1,   const float* __restrict__ b1,
             const float* __restrict__ W2,   const float* __restrict__ b2,
             const float* __restrict__ ws,   float* __restrict__ out) {
    __shared__ float sW1[16 * LW1];                  //  8,704 B
    __shared__ float stage[4][16 * LDA];             // 33,792 B

    const int tid  = threadIdx.x;
    const int wave = tid >> 5;
    const int lane = tid & 31;
    const int m    = lane & 15;
    const int kb   = (lane >> 4) << 1;
    const int rh   = (lane >= 16) ? 8 : 0;
    const int row0 = (blockIdx.x * 4 + wave) * 16;

    load_W1<128>(W1, sW1, tid);
    __syncthreads();

    // --- h' = relu(h*s + t) for this wave's 16-row tile -> LDS ---
    v8f c[8];
    gemm1_tile(diff, sW1, b1, row0, lane, c);

    const float* sArr = ws + 257;
    const float* tArr = ws + 385;
    float* st = stage[wave];
#pragma unroll
    for (int nt = 0; nt < 8; ++nt) {
        const int n = nt * 16 + m;
        const float sc = sArr[n], sh = tArr[n];
#pragma unroll
        for (int r = 0; r < 8; ++r)
            st[(r + rh) * LDA + n] = fmaxf(c[nt][r] * sc + sh, 0.f);
    }
    __syncthreads();                                 // hand tiles to all waves

    // --- GEMM2: this wave handles N-tiles {2*wave, 2*wave+1} x 64 rows ---
    const int row0b = blockIdx.x * 64;
#pragma unroll
    for (int ntl = 0; ntl < 2; ++ntl) {
        const int nt = wave * 2 + ntl;
        const int n  = nt * 16 + m;
        const float bias = b2[n];
        v8f acc[4];
#pragma unroll
        for (int sub = 0; sub < 4; ++sub)
#pragma unroll
            for (int i = 0; i < 8; ++i) acc[sub][i] = bias;

#pragma unroll
        for (int s = 0; s < 32; ++s) {
            const int k0 = 4 * s + kb;
            v2f b;                                   // one W2 frag, 4 WMMAs
            b.x = W2[k0 * DIM + n];
            b.y = W2[(k0 + 1) * DIM + n];
#pragma unroll
            for (int sub = 0; sub < 4; ++sub) {
                v2f a;                               // ds_load_b64 from stage
                a.x = stage[sub][m * LDA + k0];
                a.y = stage[sub][m * LDA + k0 + 1];
                acc[sub] = wmma_f32(a, b, acc[sub]);
            }
        }
#pragma unroll
        for (int sub = 0; sub < 4; ++sub) {
            const int rbase = row0b + sub * 16 + rh;
#pragma unroll
            for (int r = 0; r < 8; ++r) {
                const int row = rbase + r;
                const float mf = mask[row] ? 1.f : 0.f;
                out[(size_t)row * DIM + n] = acc[sub][r] * mf;
            }
        }
    }
}

// ---------------------------------------------------------------------------
// Phase 4: second tuple output = mask as 0/1 floats.
// ---------------------------------------------------------------------------
__global__ void k_mask_out(const int* __restrict__ mask, float* __restrict__ mout) {
    int i = blockIdx.x * 256 + threadIdx.x;
    if (i < M_TOK) mout[i] = mask[i] ? 1.f : 0.f;
}

// ---------------------------------------------------------------------------
extern "C" void kernel_launch(void* const* d_in, const int* in_sizes, int n_in,
                              void* d_out, int out_size, void* d_ws, size_t ws_size,
                              hipStream_t stream) {
    const float* diff  = (const float*)d_in[0];
    const int*   mask  = (const int*)  d_in[1];   // bool -> int32 per harness contract
    const float* W1    = (const float*)d_in[2];
    const float* b1    = (const float*)d_in[3];
    const float* gamma = (const float*)d_in[4];
    const float* beta  = (const float*)d_in[5];
    const float* W2    = (const float*)d_in[6];
    const float* b2    = (const float*)d_in[7];
    float*       out   = (float*)d_out;
    float*       ws    = (float*)d_ws;

    k_zero<<<1, 320, 0, stream>>>(ws);
    k_gemm1_stats<<<256, 256, 0, stream>>>(diff, mask, W1, b1, ws);
    k_finalize<<<1, 128, 0, stream>>>(gamma, beta, ws);
    k_gemm2<<<NTILES / 4, 128, 0, stream>>>(diff, mask, W1, b1, W2, b2, ws, out);
    k_mask_out<<<M_TOK / 256, 256, 0, stream>>>(mask, out + (size_t)M_TOK * DIM);
}